// DiscretePolicy_56246891708949
// MI455X (gfx1250) — compile-verified
//
#include <hip/hip_runtime.h>

typedef __attribute__((ext_vector_type(2))) float v2f;
typedef __attribute__((ext_vector_type(4))) float v4f;
typedef __attribute__((ext_vector_type(8))) float v8f;

#define NB    16384   // batch rows
#define DIN   204     // input features
#define NPAIR 100     // entity pairs per row
#define PSTR  70      // padded LDS row stride (floats): 8*70 % 64 == 48 -> no half-wave bank clash

static __device__ __forceinline__ v8f wmma4(v2f a, v2f b, v8f c) {
  // D = A(16x4 f32) * B(4x16 f32) + C(16x16 f32)
  return __builtin_amdgcn_wmma_f32_16x16x4_f32(false, a, false, b, (short)0, c, false, false);
}

static __device__ __forceinline__ v8f splat8(float v) {
  v8f r;
#pragma unroll
  for (int e = 0; e < 8; ++e) r[e] = v;
  return r;
}

static __device__ __forceinline__ void lds_fence() {
  asm volatile("s_wait_dscnt 0" ::: "memory");
}

// ---------------- Kernel 1: batch-norm stats per column -> interleaved (scale, shift) ----------------
__global__ __launch_bounds__(256) void bn_stats_kernel(const float* __restrict__ X,
                                                       float* __restrict__ nrm) {
  const int c = blockIdx.x;  // 0..DIN-1
  float s = 0.f, s2 = 0.f;
  for (int r = threadIdx.x; r < NB; r += 256) {
    float v = X[(size_t)r * DIN + c];
    s += v;
    s2 += v * v;
  }
  __shared__ float red0[256], red1[256];
  red0[threadIdx.x] = s;
  red1[threadIdx.x] = s2;
  __syncthreads();
  for (int st = 128; st > 0; st >>= 1) {
    if (threadIdx.x < st) {
      red0[threadIdx.x] += red0[threadIdx.x + st];
      red1[threadIdx.x] += red1[threadIdx.x + st];
    }
    __syncthreads();
  }
  if (threadIdx.x == 0) {
    float m   = red0[0] * (1.f / NB);
    float var = red1[0] * (1.f / NB) - m * m;   // biased variance (ddof=0)
    float rs  = rsqrtf(var + 1e-5f);
    nrm[2 * c]     = rs;        // scale
    nrm[2 * c + 1] = -m * rs;   // shift:  Xn = x*scale + shift
  }
}

// ---------------- dense layer helper: h(16xN) = act( A(16xK) @ W^T + bias ) ----------------
// A rows come from LDS polbuf (stride PSTR). W is [N][K] row-major in global (L2-resident).
template <int KSTEPS, int NT, bool RELU>
static __device__ __forceinline__ void dense_layer(const float* pb, int nc, bool hi,
                                                   const float* __restrict__ W, int wstride,
                                                   const float* __restrict__ bias, v8f* acc) {
  const int kof = hi ? 2 : 0;  // lanes 16-31 carry K-slots 2,3 of each 4-wide K step
#pragma unroll
  for (int t = 0; t < NT; ++t) acc[t] = splat8(bias[t * 16 + nc]);
#pragma unroll
  for (int kk = 0; kk < KSTEPS; ++kk) {
    const int ko = 4 * kk + kof;
    v2f a = *(const v2f*)(pb + nc * PSTR + ko);
#pragma unroll
    for (int t = 0; t < NT; ++t) {
      const int n = t * 16 + nc;
      v2f b = *(const v2f*)(W + n * wstride + ko);
      acc[t] = wmma4(a, b, acc[t]);
    }
  }
  if (RELU) {
#pragma unroll
    for (int t = 0; t < NT; ++t)
#pragma unroll
      for (int e = 0; e < 8; ++e) acc[t][e] = fmaxf(acc[t][e], 0.01f * acc[t][e]);
  }
}

static __device__ __forceinline__ void store_h(float* pb, int nc, bool hi, const v8f* h) {
#pragma unroll
  for (int t = 0; t < 4; ++t)
#pragma unroll
    for (int e = 0; e < 8; ++e) {
      const int r = e + (hi ? 8 : 0);   // C/D layout: VGPR e holds row e (lo half) / e+8 (hi half)
      pb[r * PSTR + t * 16 + nc] = h[t][e];
    }
}

// ---------------- Kernel 2: fused entity encoder + MLP + softmax ----------------
__global__ __launch_bounds__(128) void policy_fused_kernel(
    const float* __restrict__ X, const float* __restrict__ nrm,
    const float* __restrict__ Wme, const float* __restrict__ bme,
    const float* __restrict__ W1, const float* __restrict__ b1,
    const float* __restrict__ W2, const float* __restrict__ b2,
    const float* __restrict__ W3, const float* __restrict__ b3,
    float* __restrict__ out) {
  __shared__ float polbuf[4][16 * PSTR];

  const int  lane = threadIdx.x & 31;
  const int  wv   = threadIdx.x >> 5;
  const int  nc   = lane & 15;      // tile column (B/C/D) and A-matrix row index
  const bool hi   = (lane >= 16);
  const int  rowbase = blockIdx.x * 64 + wv * 16;
  const int  myrow   = rowbase + nc;
  float* pb = polbuf[wv];
  const float* xrow = X + (size_t)myrow * DIN;

  // ---- entity encoder: z[b][h] per pair k via 16x16x4 f32 WMMA (K slots 0,1 used) ----
  v2f bw[4];  // B operand: lanes<16 hold W_me rows K=0,1 ; lanes>=16 hold zeros (K=2,3)
#pragma unroll
  for (int t = 0; t < 4; ++t) {
    const int n = t * 16 + nc;
    v2f w = *(const v2f*)(Wme + 2 * n);
    v2f z; z.x = 0.f; z.y = 0.f;
    bw[t] = hi ? z : w;
  }

  v8f absacc[4];
#pragma unroll
  for (int t = 0; t < 4; ++t) absacc[t] = splat8(0.f);
  v2f asum; asum.x = 0.f; asum.y = 0.f;
  const v8f zero8 = splat8(0.f);

  // software-pipelined: loads for pair k+1 issue before the WMMA/VALU block of pair k,
  // so s_wait_kmcnt/loadcnt land after ~36 covering instructions instead of stalling.
  v2f xv = *(const v2f*)(xrow + 4);       // pair columns (x0, x1)
  v4f nv = *(const v4f*)(nrm + 8);        // (sc0, sh0, sc1, sh1) — one s_load_b128
  for (int k = 0; k < NPAIR; ++k) {
    v2f av;
    av.x = fmaf(xv.x, nv.x, nv.y);
    av.y = fmaf(xv.y, nv.z, nv.w);
    const int cn = (k < NPAIR - 1) ? (6 + 2 * k) : 4;  // next pair (clamped: no OOB on last row)
    xv = *(const v2f*)(xrow + cn);
    nv = *(const v4f*)(nrm + 2 * cn);
    v2f z2; z2.x = 0.f; z2.y = 0.f;
    av = hi ? z2 : av;                    // lanes>=16 = K slots 2,3 = zero
    asum.x += av.x; asum.y += av.y;
#pragma unroll
    for (int t = 0; t < 4; ++t) {
      v8f z = wmma4(av, bw[t], zero8);    // pre-activation for pair k (bias added analytically)
#pragma unroll
      for (int e = 0; e < 8; ++e) absacc[t][e] += fabsf(z[e]);  // v_add_f32 with |src|
    }
  }

  // mean_k lrelu(z_k) = (0.505*Sum z + 0.495*Sum|z|) / 100 ; Sum z via one WMMA, C = 100*bias
#pragma unroll
  for (int t = 0; t < 4; ++t) {
    v8f cb = splat8(100.f * bme[t * 16 + nc]);
    v8f zs = wmma4(asum, bw[t], cb);
#pragma unroll
    for (int e = 0; e < 8; ++e) {
      const float m = (0.505f * zs[e] + 0.495f * absacc[t][e]) * 0.01f;
      const int r = e + (hi ? 8 : 0);
      pb[r * PSTR + 4 + t * 16 + nc] = m;      // pol_in cols 4..67
    }
  }
  if (!hi) {
#pragma unroll
    for (int c = 0; c < 4; ++c)
      pb[nc * PSTR + c] = fmaf(xrow[c], nrm[2 * c], nrm[2 * c + 1]);  // pol_in cols 0..3
  }
  lds_fence();

  // ---- MLP: 68 -> 64 -> 64 -> 32 ----
  v8f h[4];
  dense_layer<17, 4, true>(pb, nc, hi, W1, 68, b1, h);
  lds_fence();
  store_h(pb, nc, hi, h);
  lds_fence();
  dense_layer<16, 4, true>(pb, nc, hi, W2, 64, b2, h);
  lds_fence();
  store_h(pb, nc, hi, h);
  lds_fence();
  v8f lg[2];
  dense_layer<16, 2, false>(pb, nc, hi, W3, 64, b3, lg);

  // ---- softmax over 32 logits per row; each row lives in one 16-lane half ----
#pragma unroll
  for (int r = 0; r < 8; ++r) {
    const float a0 = lg[0][r], a1 = lg[1][r];
    float m = fmaxf(a0, a1);
#pragma unroll
    for (int off = 8; off >= 1; off >>= 1) m = fmaxf(m, __shfl_xor(m, off, 32));
    const float e0 = __expf(a0 - m);
    const float e1 = __expf(a1 - m);
    float s = e0 + e1;
#pragma unroll
    for (int off = 8; off >= 1; off >>= 1) s += __shfl_xor(s, off, 32);
    const float inv = 1.0f / s;
    const int row = rowbase + r + (hi ? 8 : 0);
    out[row * 32 + nc]      = e0 * inv;
    out[row * 32 + 16 + nc] = e1 * inv;
  }
}

extern "C" void kernel_launch(void* const* d_in, const int* in_sizes, int n_in,
                              void* d_out, int out_size, void* d_ws, size_t ws_size,
                              hipStream_t stream) {
  const float* X   = (const float*)d_in[0];
  const float* Wme = (const float*)d_in[1];
  const float* bme = (const float*)d_in[2];
  const float* W1  = (const float*)d_in[3];
  const float* b1  = (const float*)d_in[4];
  const float* W2  = (const float*)d_in[5];
  const float* b2  = (const float*)d_in[6];
  const float* W3  = (const float*)d_in[7];
  const float* b3  = (const float*)d_in[8];
  float* out = (float*)d_out;

  float* nrm = (float*)d_ws;  // 2*DIN floats, interleaved (scale, shift) per column

  bn_stats_kernel<<<DIN, 256, 0, stream>>>(X, nrm);
  policy_fused_kernel<<<NB / 64, 128, 0, stream>>>(X, nrm, Wme, bme,
                                                   W1, b1, W2, b2, W3, b3, out);
}